// GIN_42872363549081
// MI455X (gfx1250) — compile-verified
//
#include <hip/hip_runtime.h>
#include <math.h>

// GIN (2x GINConv) for MI455X / gfx1250, wave32.
// Strategy: exploit linearity -> project x to H=32 BEFORE edge aggregation
// (4x less edge traffic), aggregate with native f32 L2 atomics (1 wave = 1
// edge, 32 lanes = 32 channels), and run all dense GEMMs through
// V_WMMA_F32_16X16X4_F32 (exact fp32) with fused bias/BN/ReLU epilogues.

typedef __attribute__((ext_vector_type(2))) float f32x2;
typedef __attribute__((ext_vector_type(8))) float f32x8;

#define BN_EPS 1e-5f

// Guarantee the hardware GLOBAL_ATOMIC_ADD_F32 (never a CAS loop): HIP's
// unsafeAtomicAdd maps directly to the native FP atomic on AMD targets.
__device__ __forceinline__ void atomAddF32(float* p, float v) {
    unsafeAtomicAdd(p, v);
}

// ---------------------------------------------------------------------------
// WMMA GEMM: C[nrows x NOUT] = epilogue(A (+A2) @ W + bias)
// One wave computes a 16 x NOUT tile. K, NOUT multiples of 4 / 16.
// EPI: 0=none, 1=+bias, 2=relu(+bias), 3=relu(BN(+bias))
// ---------------------------------------------------------------------------
template <int K, int NOUT, bool ADD2, int EPI>
__global__ __launch_bounds__(256) void wmma_gemm_kernel(
    const float* __restrict__ A, const float* __restrict__ A2,
    const float* __restrict__ W, const float* __restrict__ bias,
    const float* __restrict__ gamma, const float* __restrict__ beta,
    const float* __restrict__ mean, const float* __restrict__ var,
    float* __restrict__ Cout, int nrows)
{
    const int lane = threadIdx.x & 31;
    const int wave = blockIdx.x * (blockDim.x >> 5) + (threadIdx.x >> 5);
    const int row0 = wave * 16;
    if (row0 >= nrows) return;           // wave-uniform: EXEC stays all-1s

    const int lo = lane & 15;            // M (for A) / N (for B,C)
    const int hi = lane >> 4;
    const int k0 = hi * 2;               // K-pair selected by lane half

    f32x8 acc[NOUT / 16] = {};

    const float* arow  = A + (size_t)(row0 + lo) * K;
    const float* arow2 = ADD2 ? (A2 + (size_t)(row0 + lo) * K) : nullptr;

    #pragma unroll
    for (int kk = 0; kk < K; kk += 4) {
        // A-fragment (16x4 f32): v0 = A[m][kk+k0], v1 = A[m][kk+k0+1]
        f32x2 afrag;
        afrag.x = arow[kk + k0];
        afrag.y = arow[kk + k0 + 1];
        if (ADD2) {
            afrag.x += arow2[kk + k0];
            afrag.y += arow2[kk + k0 + 1];
        }
        #pragma unroll
        for (int t = 0; t < NOUT / 16; ++t) {
            // B-fragment (4x16 f32): v0 = W[kk+k0][n], v1 = W[kk+k0+1][n]
            f32x2 bfrag;
            bfrag.x = W[(size_t)(kk + k0)     * NOUT + t * 16 + lo];
            bfrag.y = W[(size_t)(kk + k0 + 1) * NOUT + t * 16 + lo];
            acc[t] = __builtin_amdgcn_wmma_f32_16x16x4_f32(
                false, afrag, false, bfrag, (short)0, acc[t], false, false);
        }
    }

    #pragma unroll
    for (int t = 0; t < NOUT / 16; ++t) {
        const int n = t * 16 + lo;
        const float b = (EPI >= 1) ? bias[n] : 0.0f;
        float sc = 1.0f, sh = 0.0f;
        if (EPI == 3) {
            const float r = rsqrtf(var[n] + BN_EPS);
            sc = r * gamma[n];
            sh = beta[n] - mean[n] * sc;
        }
        #pragma unroll
        for (int v = 0; v < 8; ++v) {
            const int rm = row0 + v + 8 * hi;   // C/D layout: row = vgpr + 8*lanehalf
            float val = acc[t][v] + b;
            if (EPI == 3) val = val * sc + sh;
            if (EPI >= 2) val = fmaxf(val, 0.0f);
            Cout[(size_t)rm * NOUT + n] = val;
        }
    }
}

// ---------------------------------------------------------------------------
// Edge scatter: agg[dst][c] += w[e] * y[src][c].  One wave per edge,
// lane = channel -> coalesced 128B gather + 32 consecutive f32 atomics (L2).
// ---------------------------------------------------------------------------
template <int C>
__global__ __launch_bounds__(256) void edge_scatter_kernel(
    const int* __restrict__ srcs, const int* __restrict__ dsts,
    const float* __restrict__ w, const float* __restrict__ y,
    float* __restrict__ agg, int nedges)
{
    const long long gtid = (long long)blockIdx.x * blockDim.x + threadIdx.x;
    const int e = (int)(gtid >> 5);
    const int c = (int)(gtid & 31);
    if (e >= nedges) return;
    const int s = srcs[e];
    const int d = dsts[e];
    const float val = w[e] * y[(size_t)s * C + c];
    atomAddF32(&agg[(size_t)d * C + c], val);
}

// h[i,c] = relu(BN(y[i,c] + a[i,c] + bias[c]))  (layer-1 post-aggregation)
__global__ __launch_bounds__(256) void bn_relu_kernel(
    const float* __restrict__ y, const float* __restrict__ a,
    const float* __restrict__ bias, const float* __restrict__ g,
    const float* __restrict__ be, const float* __restrict__ m,
    const float* __restrict__ v, float* __restrict__ h, int total)
{
    const int i = blockIdx.x * blockDim.x + threadIdx.x;
    if (i >= total) return;
    const int c = i & 31;
    float val = y[i] + a[i] + bias[c];
    const float r = rsqrtf(v[c] + BN_EPS);
    val = (val - m[c]) * r * g[c] + be[c];
    h[i] = fmaxf(val, 0.0f);
}

__global__ __launch_bounds__(256) void zero_kernel(float* __restrict__ p, long long n)
{
    long long i = (long long)blockIdx.x * blockDim.x + threadIdx.x;
    const long long stride = (long long)gridDim.x * blockDim.x;
    for (; i < n; i += stride) p[i] = 0.0f;
}

extern "C" void kernel_launch(void* const* d_in, const int* in_sizes, int n_in,
                              void* d_out, int out_size, void* d_ws, size_t ws_size,
                              hipStream_t stream)
{
    const float* x   = (const float*)d_in[0];
    const float* ew  = (const float*)d_in[1];
    const float* W1a = (const float*)d_in[2];
    const float* b1a = (const float*)d_in[3];
    const float* g1  = (const float*)d_in[4];
    const float* be1 = (const float*)d_in[5];
    const float* m1  = (const float*)d_in[6];
    const float* v1  = (const float*)d_in[7];
    const float* W1b = (const float*)d_in[8];
    const float* b1b = (const float*)d_in[9];
    const float* W2a = (const float*)d_in[10];
    const float* b2a = (const float*)d_in[11];
    const float* g2  = (const float*)d_in[12];
    const float* be2 = (const float*)d_in[13];
    const float* m2  = (const float*)d_in[14];
    const float* v2  = (const float*)d_in[15];
    const float* W2b = (const float*)d_in[16];
    const float* b2b = (const float*)d_in[17];
    const int*   eix = (const int*)d_in[18];

    const int N = in_sizes[0] / 128;     // 100000 (divisible by 16)
    const int E = in_sizes[18] / 2;      // 1600000
    const int* esrc = eix;
    const int* edst = eix + E;

    // workspace layout (floats), total N*128: [bufA][bufB][bufC][bufD]
    //   bufA: y1 = x@W1a, later reused as a2
    //   bufB: a1 ; later h2 (N x 64) overlays bufB+bufC
    //   bufC: h1 ; bufD: o1
    float* ws   = (float*)d_ws;
    float* bufA = ws;
    float* bufB = ws + (size_t)N * 32;
    float* bufC = ws + (size_t)N * 64;
    float* bufD = ws + (size_t)N * 96;
    float* h2   = bufB;                  // N x 64, overlays dead a1+h1
    float* out  = (float*)d_out;

    const int gemmWaves  = N / 16;
    const int gemmBlocks = (gemmWaves + 7) / 8;              // 8 waves / block
    const int edgeBlocks = (E * 32 + 255) / 256;
    const int nodeBlocks = (N * 32 + 255) / 256;

    // ---- Layer 1 ----
    zero_kernel<<<2048, 256, 0, stream>>>(bufB, (long long)N * 32);   // a1 = 0
    // y1 = x @ W1a   (project BEFORE aggregation: GIN agg is linear)
    wmma_gemm_kernel<128, 32, false, 0><<<gemmBlocks, 256, 0, stream>>>(
        x, nullptr, W1a, nullptr, nullptr, nullptr, nullptr, nullptr, bufA, N);
    // a1[dst] += w_e * y1[src]
    edge_scatter_kernel<32><<<edgeBlocks, 256, 0, stream>>>(esrc, edst, ew, bufA, bufB, E);
    // h1 = relu(BN1(y1 + a1 + b1a))
    bn_relu_kernel<<<nodeBlocks, 256, 0, stream>>>(bufA, bufB, b1a, g1, be1, m1, v1, bufC, N * 32);
    // o1 = relu(h1 @ W1b + b1b)   (includes outer relu of layer 1)
    wmma_gemm_kernel<32, 32, false, 2><<<gemmBlocks, 256, 0, stream>>>(
        bufC, nullptr, W1b, b1b, nullptr, nullptr, nullptr, nullptr, bufD, N);

    // ---- Layer 2 ----
    zero_kernel<<<2048, 256, 0, stream>>>(bufA, (long long)N * 32);   // a2 = 0
    // a2[dst] += w_e * o1[src]
    edge_scatter_kernel<32><<<edgeBlocks, 256, 0, stream>>>(esrc, edst, ew, bufD, bufA, E);
    // h2 = relu(BN2((o1 + a2) @ W2a + b2a))   (sum fused into A-fragment load)
    wmma_gemm_kernel<32, 64, true, 3><<<gemmBlocks, 256, 0, stream>>>(
        bufD, bufA, W2a, b2a, g2, be2, m2, v2, h2, N);
    // out = h2 @ W2b + b2b
    wmma_gemm_kernel<64, 64, false, 1><<<gemmBlocks, 256, 0, stream>>>(
        h2, nullptr, W2b, b2b, nullptr, nullptr, nullptr, nullptr, out, N);
}